// TargetedDropout_22136261443661
// MI455X (gfx1250) — compile-verified
//
#include <hip/hip_runtime.h>

// TargetedDropout (inference / pruned path) for x:(16,2048,1024) fp32.
// Per channel c (last axis), threshold = R-th smallest |x| over the N=32768
// leading elements (R = N/2); output zeroes every element with |x| <= thr[c].
//
// Exact 3-level radix select on abs-bit patterns (monotonic as u32):
//   L0: bits[30:20] (2048 bins) -> L1: bits[19:10] (1024) -> L2: bits[9:0] (1024)
// x (134MB) fits in the 192MB L2, so refine passes stream from L2.
// Scan kernels re-zero the histogram in place for the next level.

#define C_CH   1024
#define BINS0  2048
#define BINS1  1024
#define BINS2  1024

// Native clang vector (required by __builtin_nontemporal_store).
typedef __attribute__((ext_vector_type(4))) float fvec4;

// ---------------------------------------------------------------------------
// Async-copy staging of a 1024-entry u32 table (4KB) into LDS.
// gfx1250 path: global_load_async_to_lds_b128 (GVS mode: SGPR base + VGPR
// byte offset; VDST VGPR holds the LDS byte address), tracked by ASYNCcnt.
// ---------------------------------------------------------------------------
__device__ __forceinline__ void async_stage_1k(const unsigned* __restrict__ gsrc,
                                               unsigned* __restrict__ lds) {
  unsigned t = threadIdx.x;
  if (t < 256u) {
    // Low 32 bits of a generic pointer into LDS are the LDS byte address.
    unsigned lds_off = (unsigned)(unsigned long long)(&lds[t * 4u]);
    unsigned goff    = t * 16u;  // 16 bytes per lane, 256 lanes -> 4KB
    asm volatile("global_load_async_to_lds_b128 %0, %1, %2"
                 :: "v"(lds_off), "v"(goff), "s"(gsrc)
                 : "memory");
  }
  asm volatile("s_wait_asynccnt 0" ::: "memory");
  __syncthreads();
}

// ---------------------------------------------------------------------------
// Zero the histogram workspace (uint4 stores, 16B per lane). Runs once per
// launch (d_ws is poisoned by the harness); later re-zeroing is fused into
// the scan kernels.
// ---------------------------------------------------------------------------
__global__ void td_zero_kernel(uint4* __restrict__ p, int n4) {
  int i = blockIdx.x * blockDim.x + threadIdx.x;
  int stride = gridDim.x * blockDim.x;
  uint4 z; z.x = 0u; z.y = 0u; z.z = 0u; z.w = 0u;
  for (; i < n4; i += stride) p[i] = z;
}

// ---------------------------------------------------------------------------
// Level-0 histogram: bin = abs_bits >> 20 (2048 bins), per channel.
// ---------------------------------------------------------------------------
__global__ void td_hist0_kernel(const float4* __restrict__ x4,
                                unsigned* __restrict__ hist, int n4) {
  int i = blockIdx.x * blockDim.x + threadIdx.x;
  int stride = gridDim.x * blockDim.x;
  for (; i < n4; i += stride) {
    __builtin_prefetch((const void*)(x4 + i + stride), 0, 3);
    float4 v = x4[i];
    unsigned c0 = ((unsigned)i << 2) & (C_CH - 1u);
    unsigned u;
    u = __float_as_uint(v.x) & 0x7FFFFFFFu; atomicAdd(&hist[(u >> 20) * C_CH + c0 + 0u], 1u);
    u = __float_as_uint(v.y) & 0x7FFFFFFFu; atomicAdd(&hist[(u >> 20) * C_CH + c0 + 1u], 1u);
    u = __float_as_uint(v.z) & 0x7FFFFFFFu; atomicAdd(&hist[(u >> 20) * C_CH + c0 + 2u], 1u);
    u = __float_as_uint(v.w) & 0x7FFFFFFFu; atomicAdd(&hist[(u >> 20) * C_CH + c0 + 3u], 1u);
  }
}

// ---------------------------------------------------------------------------
// Level-0 scan: per channel, find bin containing rank R; residual rank out.
// hist layout hist[bin*C + c] -> loads coalesced across lanes (lane=channel).
// Zeroes each cell after reading (prepares hist for level 1).
// ---------------------------------------------------------------------------
__global__ void td_scan0_kernel(unsigned* __restrict__ hist,
                                unsigned* __restrict__ sel0,
                                unsigned* __restrict__ rank1, unsigned R) {
  int c = blockIdx.x * blockDim.x + threadIdx.x;
  if (c >= C_CH) return;
  unsigned cum = 0u, sel = 0u, rk = 0u;
  bool found = false;
  for (int b = 0; b < BINS0; ++b) {
    unsigned h = hist[b * C_CH + c];
    hist[b * C_CH + c] = 0u;  // fused re-zero for next level
    if (!found && (cum + h >= R)) { sel = (unsigned)b; rk = R - cum; found = true; }
    cum += h;
  }
  sel0[c]  = sel;
  rank1[c] = rk;
}

// ---------------------------------------------------------------------------
// Level-1 histogram: only elements whose bits[30:20] match sel0[c];
// bin = bits[19:10] (1024 bins). sel0 table staged via async-to-LDS.
// ---------------------------------------------------------------------------
__global__ void td_hist1_kernel(const float4* __restrict__ x4,
                                const unsigned* __restrict__ sel0,
                                unsigned* __restrict__ hist, int n4) {
  __shared__ __align__(16) unsigned s_sel[C_CH];
  async_stage_1k(sel0, s_sel);

  int i = blockIdx.x * blockDim.x + threadIdx.x;
  int stride = gridDim.x * blockDim.x;
  for (; i < n4; i += stride) {
    __builtin_prefetch((const void*)(x4 + i + stride), 0, 3);
    float4 v = x4[i];
    unsigned c0 = ((unsigned)i << 2) & (C_CH - 1u);
    unsigned u;
    u = __float_as_uint(v.x) & 0x7FFFFFFFu;
    if ((u >> 20) == s_sel[c0 + 0u]) atomicAdd(&hist[((u >> 10) & 0x3FFu) * C_CH + c0 + 0u], 1u);
    u = __float_as_uint(v.y) & 0x7FFFFFFFu;
    if ((u >> 20) == s_sel[c0 + 1u]) atomicAdd(&hist[((u >> 10) & 0x3FFu) * C_CH + c0 + 1u], 1u);
    u = __float_as_uint(v.z) & 0x7FFFFFFFu;
    if ((u >> 20) == s_sel[c0 + 2u]) atomicAdd(&hist[((u >> 10) & 0x3FFu) * C_CH + c0 + 2u], 1u);
    u = __float_as_uint(v.w) & 0x7FFFFFFFu;
    if ((u >> 20) == s_sel[c0 + 3u]) atomicAdd(&hist[((u >> 10) & 0x3FFu) * C_CH + c0 + 3u], 1u);
  }
}

// ---------------------------------------------------------------------------
// Level-1 scan: prefix21 = (sel0 << 10) | sel1 ; residual rank out.
// Zeroes each cell after reading (prepares hist for level 2).
// ---------------------------------------------------------------------------
__global__ void td_scan1_kernel(unsigned* __restrict__ hist,
                                const unsigned* __restrict__ sel0,
                                const unsigned* __restrict__ rank1,
                                unsigned* __restrict__ pref21,
                                unsigned* __restrict__ rank2) {
  int c = blockIdx.x * blockDim.x + threadIdx.x;
  if (c >= C_CH) return;
  unsigned R1 = rank1[c];
  unsigned cum = 0u, sel = 0u, rk = 0u;
  bool found = false;
  for (int b = 0; b < BINS1; ++b) {
    unsigned h = hist[b * C_CH + c];
    hist[b * C_CH + c] = 0u;  // fused re-zero for next level
    if (!found && (cum + h >= R1)) { sel = (unsigned)b; rk = R1 - cum; found = true; }
    cum += h;
  }
  pref21[c] = (sel0[c] << 10) | sel;
  rank2[c]  = rk;
}

// ---------------------------------------------------------------------------
// Level-2 histogram: elements whose bits[30:10] match pref21[c];
// bin = bits[9:0].
// ---------------------------------------------------------------------------
__global__ void td_hist2_kernel(const float4* __restrict__ x4,
                                const unsigned* __restrict__ pref21,
                                unsigned* __restrict__ hist, int n4) {
  __shared__ __align__(16) unsigned s_pref[C_CH];
  async_stage_1k(pref21, s_pref);

  int i = blockIdx.x * blockDim.x + threadIdx.x;
  int stride = gridDim.x * blockDim.x;
  for (; i < n4; i += stride) {
    __builtin_prefetch((const void*)(x4 + i + stride), 0, 3);
    float4 v = x4[i];
    unsigned c0 = ((unsigned)i << 2) & (C_CH - 1u);
    unsigned u;
    u = __float_as_uint(v.x) & 0x7FFFFFFFu;
    if ((u >> 10) == s_pref[c0 + 0u]) atomicAdd(&hist[(u & 0x3FFu) * C_CH + c0 + 0u], 1u);
    u = __float_as_uint(v.y) & 0x7FFFFFFFu;
    if ((u >> 10) == s_pref[c0 + 1u]) atomicAdd(&hist[(u & 0x3FFu) * C_CH + c0 + 1u], 1u);
    u = __float_as_uint(v.z) & 0x7FFFFFFFu;
    if ((u >> 10) == s_pref[c0 + 2u]) atomicAdd(&hist[(u & 0x3FFu) * C_CH + c0 + 2u], 1u);
    u = __float_as_uint(v.w) & 0x7FFFFFFFu;
    if ((u >> 10) == s_pref[c0 + 3u]) atomicAdd(&hist[(u & 0x3FFu) * C_CH + c0 + 3u], 1u);
  }
}

// ---------------------------------------------------------------------------
// Level-2 scan: exact threshold abs-bits = (pref21 << 10) | sel2.
// ---------------------------------------------------------------------------
__global__ void td_scan2_kernel(const unsigned* __restrict__ hist,
                                const unsigned* __restrict__ pref21,
                                const unsigned* __restrict__ rank2,
                                unsigned* __restrict__ thr) {
  int c = blockIdx.x * blockDim.x + threadIdx.x;
  if (c >= C_CH) return;
  unsigned R2 = rank2[c];
  unsigned cum = 0u, sel = 0u;
  bool found = false;
  for (int b = 0; b < BINS2; ++b) {
    unsigned h = hist[b * C_CH + c];
    if (!found && (cum + h >= R2)) { sel = (unsigned)b; found = true; }
    cum += h;
  }
  thr[c] = (pref21[c] << 10) | sel;
}

// ---------------------------------------------------------------------------
// Mask pass: out = (|x| <= thr[c]) ? 0 : x, via unsigned compare on abs bits
// (exact match to reference semantics, including ties). thr staged via async.
// Output stored nontemporal (never re-read; keep x's L2 lines intact).
// Uses native ext_vector fvec4 so __builtin_nontemporal_store accepts it.
// ---------------------------------------------------------------------------
__global__ void td_mask_kernel(const fvec4* __restrict__ x4,
                               const unsigned* __restrict__ thr,
                               fvec4* __restrict__ out4, int n4) {
  __shared__ __align__(16) unsigned s_thr[C_CH];
  async_stage_1k(thr, s_thr);

  int i = blockIdx.x * blockDim.x + threadIdx.x;
  int stride = gridDim.x * blockDim.x;
  for (; i < n4; i += stride) {
    __builtin_prefetch((const void*)(x4 + i + stride), 0, 3);
    fvec4 v = x4[i];
    unsigned c0 = ((unsigned)i << 2) & (C_CH - 1u);
    unsigned u;
    u = __float_as_uint(v.x) & 0x7FFFFFFFu; if (u <= s_thr[c0 + 0u]) v.x = 0.0f;
    u = __float_as_uint(v.y) & 0x7FFFFFFFu; if (u <= s_thr[c0 + 1u]) v.y = 0.0f;
    u = __float_as_uint(v.z) & 0x7FFFFFFFu; if (u <= s_thr[c0 + 2u]) v.z = 0.0f;
    u = __float_as_uint(v.w) & 0x7FFFFFFFu; if (u <= s_thr[c0 + 3u]) v.w = 0.0f;
    __builtin_nontemporal_store(v, &out4[i]);
  }
}

// ---------------------------------------------------------------------------
extern "C" void kernel_launch(void* const* d_in, const int* in_sizes, int n_in,
                              void* d_out, int out_size, void* d_ws, size_t ws_size,
                              hipStream_t stream) {
  (void)n_in; (void)out_size; (void)ws_size;

  const float* x = (const float*)d_in[0];
  float* out = (float*)d_out;

  const int total = in_sizes[0];                   // 16*2048*1024 = 33,554,432
  const int n4 = total / 4;
  const unsigned nrow = (unsigned)(total / C_CH);  // 32768
  const unsigned R = nrow / 2u;                    // 1-based rank of threshold

  // Workspace layout (~8.02 MB): one histogram buffer reused across levels.
  unsigned* hist   = (unsigned*)d_ws;              // BINS0 * C_CH u32 = 8MB
  unsigned* sel0   = hist + (size_t)BINS0 * C_CH;
  unsigned* rank1  = sel0 + C_CH;
  unsigned* pref21 = rank1 + C_CH;
  unsigned* rank2  = pref21 + C_CH;
  unsigned* thr    = rank2 + C_CH;

  const dim3 blk(256);                             // 8 wave32 waves per block
  const int histBlocks = 4096;                     // grid-stride over 8.4M float4
  const int scanBlocks = C_CH / 256;               // 4

  const int zero0_n4 = (BINS0 * C_CH) / 4;

  // Initial zero only (d_ws is poisoned once by the harness; scans re-zero).
  hipLaunchKernelGGL(td_zero_kernel, dim3((zero0_n4 + 255) / 256), blk, 0, stream,
                     (uint4*)hist, zero0_n4);
  // Level 0
  hipLaunchKernelGGL(td_hist0_kernel, dim3(histBlocks), blk, 0, stream,
                     (const float4*)x, hist, n4);
  hipLaunchKernelGGL(td_scan0_kernel, dim3(scanBlocks), blk, 0, stream,
                     hist, sel0, rank1, R);
  // Level 1
  hipLaunchKernelGGL(td_hist1_kernel, dim3(histBlocks), blk, 0, stream,
                     (const float4*)x, sel0, hist, n4);
  hipLaunchKernelGGL(td_scan1_kernel, dim3(scanBlocks), blk, 0, stream,
                     hist, sel0, rank1, pref21, rank2);
  // Level 2
  hipLaunchKernelGGL(td_hist2_kernel, dim3(histBlocks), blk, 0, stream,
                     (const float4*)x, pref21, hist, n4);
  hipLaunchKernelGGL(td_scan2_kernel, dim3(scanBlocks), blk, 0, stream,
                     hist, pref21, rank2, thr);
  // Mask
  hipLaunchKernelGGL(td_mask_kernel, dim3(histBlocks), blk, 0, stream,
                     (const fvec4*)x, thr, (fvec4*)out, n4);
}